// cMSE_48945447305596
// MI455X (gfx1250) — compile-verified
//
#include <hip/hip_runtime.h>

// ---------------------------------------------------------------------------
// cMSE (masked, bias-corrected, best-of-49-shifts) for MI455X / gfx1250.
// Core correlation is a skinny GEMM D = G x F^T done with
// V_WMMA_F32_16X16X4_F32 (f32 exact, matches f32 reference precision).
//
// Row permutation (147 rows in 10 tiles of 16) chosen so 9/10 tiles are
// quantity-pure (no per-lane selects, and the mask-only tiles skip hr loads):
//   tiles 0-2 : q=0 (mask)      s = t*16+ln        (s 0..47)
//   tile  3   : ln=0/1/2 -> q=0/1/2 with s=48 ; ln>=3 pad (q=2,s=0, ignored)
//   tiles 4-6 : q=1 (L=hr*mask) s = (t-4)*16+ln    (s 0..47)
//   tiles 7-9 : q=2 (L^2)       s = (t-7)*16+ln    (s 0..47)
//   F cols (N=3 of 16): {1, pred, pred^2},  pred = (sr*mask)[3:381,3:381]
// Chunk loads use an SGPR base (wave-uniform index via readfirstlane) +
// loop-invariant VGPR offsets, batch-issued before the WMMA group.
// ---------------------------------------------------------------------------

typedef __attribute__((ext_vector_type(2))) float v2f;
typedef __attribute__((ext_vector_type(8))) float v8f;

#define HH      384
#define CROP    378
#define NT      10
#define WS_PER_B 512        // floats per batch: [(q*49+s)*3+col ... , P at 480]
#define ROWS_PER_WAVE 7
#define WAVES_PER_BLK 8     // 256 threads, wave32
#define GROUPS_PER_B  7     // ceil(378 / (8*7)) = 7

__global__ void zero_ws_kernel(float* ws, int n) {
    int i = blockIdx.x * blockDim.x + threadIdx.x;
    if (i < n) ws[i] = 0.0f;
}

// Tile/row -> (quantity, shift) permutation; must match between GEMM and reduce.
__device__ __forceinline__ void rowmap(int t, int m, int& q, int& s, bool& valid) {
    if (t < 3)       { q = 0;                 s = t * 16 + m;       valid = true; }
    else if (t == 3) { q = (m < 3) ? m : 2;   s = (m < 3) ? 48 : 0; valid = (m < 3); }
    else if (t < 7)  { q = 1;                 s = (t - 4) * 16 + m; valid = true; }
    else             { q = 2;                 s = (t - 7) * 16 + m; valid = true; }
}

__global__ __launch_bounds__(256)
void corr_wmma_kernel(const float* __restrict__ sr,
                      const float* __restrict__ hr,
                      const float* __restrict__ mk,
                      float* __restrict__ ws)
{
    const int b    = blockIdx.y;
    const int wave = threadIdx.x >> 5;
    const int lane = threadIdx.x & 31;
    const int ln   = lane & 15;     // M-row within tile (A) / N-col (B)
    const int half = lane >> 4;     // k-pair selector: klo = 2*half
    const int klo  = half << 1;

    const size_t ib = (size_t)b * (HH * HH);
    const float* __restrict__ srb = sr + ib;
    const float* __restrict__ hrb = hr + ib;
    const float* __restrict__ mkb = mk + ib;

    // Loop-invariant per-lane element offsets per tile.
    int offA[NT];
#pragma unroll
    for (int t = 0; t < NT; ++t) {
        int q, s; bool valid;
        rowmap(t, ln, q, s, valid);
        int i = s / 7;
        int j = s - 7 * i;
        offA[t] = i * HH + j + klo;
    }
    const int offP = 3 * HH + 3 + klo;        // pred lives at (y+3, x+3)

    v8f acc[NT];
#pragma unroll
    for (int t = 0; t < NT; ++t) acc[t] = (v8f){0,0,0,0,0,0,0,0};
    float accP = 0.0f;                        // 16x over-counted sum of pred

    const int y0 = (blockIdx.x * WAVES_PER_BLK + wave) * ROWS_PER_WAVE;

    for (int yy = 0; yy < ROWS_PER_WAVE; ++yy) {
        const int y = y0 + yy;
        if (y >= CROP) break;                 // wave-uniform
        for (int c = 0; c < 95; ++c) {
            // chunks 0..93: x0=4c (x=0..375); chunk 94: x0=374, k=0,1 dups -> zeroed in B
            const int  x0 = (c < 94) ? (c << 2) : 374;
            const bool zp = (c == 94) && (half == 0);

            // Wave-uniform chunk base -> SGPR; loads become saddr + const voffset.
            const int idxs = __builtin_amdgcn_readfirstlane(y * HH + x0);
            const float* __restrict__ bm = mkb + idxs;
            const float* __restrict__ bh = hrb + idxs;
            const float* __restrict__ bs = srb + idxs;

            // ---- batch-issue ALL loads for this chunk before any compute ----
            float m0[NT], m1[NT], h0[NT], h1[NT];
#pragma unroll
            for (int t = 0; t < NT; ++t) {
                m0[t] = bm[offA[t]];
                m1[t] = bm[offA[t] + 1];
                if (t >= 3) {                 // mask-only tiles never touch hr
                    h0[t] = bh[offA[t]];
                    h1[t] = bh[offA[t] + 1];
                }
            }
            float s0 = bs[offP], s1 = bs[offP + 1];
            float g0 = bm[offP], g1 = bm[offP + 1];

            // ---- B fragment: cols {1, pred, pred^2} at this lane's k-pair ----
            float p0 = zp ? 0.0f : s0 * g0;
            float p1 = zp ? 0.0f : s1 * g1;
            float one = zp ? 0.0f : 1.0f;
            accP += p0 + p1;
            float b0 = (ln == 0) ? one : (ln == 1) ? p0 : (ln == 2) ? p0 * p0 : 0.0f;
            float b1 = (ln == 0) ? one : (ln == 1) ? p1 : (ln == 2) ? p1 * p1 : 0.0f;
            v2f Bf = {b0, b1};

            // ---- A fragments + 10 accumulating WMMAs ----
#pragma unroll
            for (int t = 0; t < NT; ++t) {
                float a0, a1;
                if (t < 3) {                      // pure mask tiles
                    a0 = m0[t]; a1 = m1[t];
                } else if (t == 3) {              // mixed leftover tile (s=48 rows)
                    float l0 = h0[t] * m0[t], l1 = h1[t] * m1[t];
                    float e0 = l0 * l0,       e1 = l1 * l1;
                    a0 = (ln == 0) ? m0[t] : (ln == 1) ? l0 : e0;
                    a1 = (ln == 0) ? m1[t] : (ln == 1) ? l1 : e1;
                } else if (t < 7) {               // pure L tiles
                    a0 = h0[t] * m0[t]; a1 = h1[t] * m1[t];
                } else {                          // pure L^2 tiles
                    float l0 = h0[t] * m0[t], l1 = h1[t] * m1[t];
                    a0 = l0 * l0; a1 = l1 * l1;
                }
                v2f Af = {a0, a1};
                // D(16x16,f32) += A(16x4,f32) x B(4x16,f32)
                acc[t] = __builtin_amdgcn_wmma_f32_16x16x4_f32(
                    false, Af, false, Bf, (short)0, acc[t], false, false);
            }
        }
    }

    // ---- reduce fragments across the 8 waves via LDS float atomics ----
    __shared__ float lred[488];
    for (int tt = threadIdx.x; tt < 488; tt += 256) lred[tt] = 0.0f;
    __syncthreads();

    // C/D layout: lane<16 -> N=ln, M=v ; lane>=16 -> N=ln, M=v+8 (per VGPR v)
    if (ln < 3) {
#pragma unroll
        for (int t = 0; t < NT; ++t) {
#pragma unroll
            for (int v = 0; v < 8; ++v) {
                int M = v + half * 8;
                int q, s; bool valid;
                rowmap(t, M, q, s, valid);
                if (valid) atomicAdd(&lred[(q * 49 + s) * 3 + ln], acc[t][v]);
            }
        }
    }
    atomicAdd(&lred[480], accP);
    __syncthreads();

    float* wsb = ws + (size_t)b * WS_PER_B;
    for (int tt = threadIdx.x; tt < 481; tt += 256)
        atomicAdd(&wsb[tt], lred[tt]);
}

__global__ void finalize_kernel(const float* __restrict__ ws, float* __restrict__ out, int B)
{
    __shared__ float red[256];
    const int b = threadIdx.x;
    float best = 0.0f;
    if (b < B) {
        const float* w = ws + (size_t)b * WS_PER_B;
        const float P = w[480] * (1.0f / 16.0f);   // accP was 16x over-counted
        best = 3.4e38f;
        for (int s = 0; s < 49; ++s) {
            float n    = w[(0 * 49 + s) * 3 + 0];  // sum m_s
            float SmP  = w[(0 * 49 + s) * 3 + 1];  // sum m_s * pred
            float SmP2 = w[(0 * 49 + s) * 3 + 2];  // sum m_s * pred^2
            float A    = w[(1 * 49 + s) * 3 + 0];  // sum L_s
            float SLP  = w[(1 * 49 + s) * 3 + 1];  // sum L_s * pred
            float SL2  = w[(2 * 49 + s) * 3 + 0];  // sum L_s^2
            float bias = (A - P) / n;
            float T1   = A - SmP;
            float T2   = SL2 - 2.0f * SLP + SmP2;
            float mse  = (T2 - 2.0f * bias * T1 + bias * bias * n) / n;
            best = fminf(best, mse);
        }
    }
    red[b] = (b < B) ? best : 0.0f;
    __syncthreads();
    for (int off = 128; off > 0; off >>= 1) {
        if (b < off && (b + off) < 256) red[b] += red[b + off];
        __syncthreads();
    }
    if (b == 0) out[0] = red[0] / (float)B;
}

extern "C" void kernel_launch(void* const* d_in, const int* in_sizes, int n_in,
                              void* d_out, int out_size, void* d_ws, size_t ws_size,
                              hipStream_t stream)
{
    const float* sr = (const float*)d_in[0];
    const float* hr = (const float*)d_in[1];
    const float* mk = (const float*)d_in[2];
    float* out = (float*)d_out;
    float* ws  = (float*)d_ws;

    const int B = in_sizes[0] / (HH * HH);   // 64

    const int nzero = B * WS_PER_B;
    zero_ws_kernel<<<(nzero + 255) / 256, 256, 0, stream>>>(ws, nzero);

    dim3 grid(GROUPS_PER_B, B);
    corr_wmma_kernel<<<grid, 256, 0, stream>>>(sr, hr, mk, ws);

    finalize_kernel<<<1, 256, 0, stream>>>(ws, out, B);
}